// GraphEncoder_39256001086161
// MI455X (gfx1250) — compile-verified
//
#include <hip/hip_runtime.h>
#include <math.h>

// ---- problem constants (match reference) ----
#define Nn   32768
#define Ee   524288
#define Bb   32
#define NPGc 1024
#define KKc  512
#define EMBc 64
#define HCc  128

// ---- flat param offsets (jax tree_leaves order: dict keys sorted) ----
#define BN1_B 0
#define BN1_G 64
#define BN2_B 128
#define BN2_G 192
#define C1_WE 256
#define C1_WK 4352
#define C1_WQ 20736
#define C1_WSKIP 37120
#define C1_WV 53504
#define C1_WWL 69888
#define C1_BK 86272
#define C1_BQ 86400
#define C1_BSKIP 86528
#define C1_BV 86656
#define C1_BWL 86784
#define C2_WE 86912
#define C2_WK 91008
#define C2_WQ 99200
#define C2_WSKIP 107392
#define C2_WV 115584
#define C2_WWL 123776
#define C2_BK 131968
#define C2_BQ 132096
#define C2_BSKIP 132224
#define C2_BV 132352
#define C2_BWL 132480
#define POOL_W 132608
#define T1_W 132672
#define T1_B 140864
#define T2_W 140928
#define T2_B 149120

#define ENC_NEG_INF 0x007FFFFFu   // order-preserving encoding of -inf

typedef __attribute__((ext_vector_type(2))) float v2f;
typedef __attribute__((ext_vector_type(8))) float v8f;

// ---------------- utility fills ----------------
__global__ void k_fill_f32(float* p, float v, int n) {
  int i = blockIdx.x * blockDim.x + threadIdx.x;
  if (i < n) p[i] = v;
}
__global__ void k_fill_u32(unsigned* p, unsigned v, int n) {
  int i = blockIdx.x * blockDim.x + threadIdx.x;
  if (i < n) p[i] = v;
}

// ---------------- degree / normalization ----------------
__global__ void k_deg(const int* __restrict__ src, float* deg, int n) {
  int i = blockIdx.x * blockDim.x + threadIdx.x;
  if (i < n) atomicAdd(&deg[src[i]], 1.0f);
}
__global__ void k_dis(const float* __restrict__ deg, float* dis, int n) {
  int i = blockIdx.x * blockDim.x + threadIdx.x;
  if (i < n) { float d = deg[i]; dis[i] = d > 0.0f ? rsqrtf(d) : 0.0f; }
}

// ---------------- fp32 WMMA GEMM: C[M,Nc] = A[M,Ka] @ W[Ka,Nc] (+bias) ----------------
// one wave computes one 16x16 tile via V_WMMA_F32_16X16X4_F32 chain over K.
// flags: bit0 = accumulate into C, bit1 = relu
__global__ void k_gemm_wmma(const float* __restrict__ A, const float* __restrict__ W,
                            const float* __restrict__ bias, float* __restrict__ C,
                            int M, int Ka, int Nc, int flags) {
  int wave = threadIdx.x >> 5;
  int lane = threadIdx.x & 31;
  int tM = (blockIdx.y * 8 + wave) * 16;
  int tN = blockIdx.x * 16;
  if (tM >= M) return;                       // wave-uniform; EXEC stays full for WMMA
  int r  = lane & 15;
  int kh = (lane >> 4) << 1;                 // K-half * 2
  v8f acc = {};
  const float* ap = A + (size_t)(tM + r) * Ka + kh;
  const float* bp = W + (size_t)kh * Nc + tN + r;
  for (int k0 = 0; k0 < Ka; k0 += 4) {
    v2f av, bv;
    av.x = ap[0]; av.y = ap[1];
    bv.x = bp[0]; bv.y = bp[Nc];
    acc = __builtin_amdgcn_wmma_f32_16x16x4_f32(false, av, false, bv, (short)0, acc,
                                                false, false);
    ap += 4;
    bp += 4 * Nc;
  }
  int col = tN + r;
  float bb = bias ? bias[col] : 0.0f;
  int rowBase = tM + ((lane >> 4) << 3);
#pragma unroll
  for (int j = 0; j < 8; ++j) {
    float v = acc[j] + bb;
    if (flags & 2) v = fmaxf(v, 0.0f);
    size_t idx = (size_t)(rowBase + j) * Nc + col;
    if (flags & 1) C[idx] += v; else C[idx] = v;
  }
}

// ---------------- edge pass 1: attention logits + segment max ----------------
// one wave per edge; lane handles dims d = lane*4..lane*4+3; lane halves = heads
__global__ void k_edge_alpha(const int* __restrict__ src, const int* __restrict__ dst,
                             const float* __restrict__ ea, const float* __restrict__ We,
                             const float* __restrict__ Q, const float* __restrict__ Kf,
                             float* __restrict__ alpha, unsigned* __restrict__ mEnc, int E) {
  int gw = blockIdx.x * 8 + (threadIdx.x >> 5);
  int lane = threadIdx.x & 31;
  if (gw >= E) return;
  int s = src[gw], d = dst[gw];
  float e0 = 0.f, e1 = 0.f, e2 = 0.f, e3 = 0.f;
  const float* eap = ea + (size_t)gw * 32;
#pragma unroll
  for (int f = 0; f < 32; ++f) {
    float a = eap[f];
    const float* w = We + f * HCc + lane * 4;
    e0 += a * w[0]; e1 += a * w[1]; e2 += a * w[2]; e3 += a * w[3];
  }
  const float* qp = Q + (size_t)d * HCc + lane * 4;
  const float* kp = Kf + (size_t)s * HCc + lane * 4;
  float part = qp[0] * (kp[0] + e0) + qp[1] * (kp[1] + e1) +
               qp[2] * (kp[2] + e2) + qp[3] * (kp[3] + e3);
  part += __shfl_xor(part, 1, 32);
  part += __shfl_xor(part, 2, 32);
  part += __shfl_xor(part, 4, 32);
  part += __shfl_xor(part, 8, 32);           // full sum within each 16-lane head group
  float a = part * 0.125f;                   // 1/sqrt(EMB=64)
  if ((lane & 15) == 0) {
    int h = lane >> 4;
    alpha[(size_t)gw * 2 + h] = a;
    unsigned u = __float_as_uint(a);
    u = (u & 0x80000000u) ? ~u : (u | 0x80000000u);
    atomicMax(mEnc + (size_t)d * 2 + h, u);
  }
}

// ---------------- edge pass 2: exp(alpha - max) + segment sum ----------------
__global__ void k_edge_exp(const int* __restrict__ dst, float* __restrict__ alpha,
                           const unsigned* __restrict__ mEnc, float* __restrict__ denom, int E) {
  int e = blockIdx.x * blockDim.x + threadIdx.x;
  if (e >= E) return;
  int d = dst[e];
#pragma unroll
  for (int h = 0; h < 2; ++h) {
    unsigned u = mEnc[(size_t)d * 2 + h];
    float m = (u & 0x80000000u) ? __uint_as_float(u ^ 0x80000000u) : __uint_as_float(~u);
    float a = expf(alpha[(size_t)e * 2 + h] - m);
    alpha[(size_t)e * 2 + h] = a;
    atomicAdd(&denom[(size_t)d * 2 + h], a);
  }
}

// ---------------- edge pass 3: attention scatter + degree-normalized aggregation ----------------
__global__ void k_edge_scatter(const int* __restrict__ src, const int* __restrict__ dst,
                               const float* __restrict__ ea, const float* __restrict__ We,
                               const float* __restrict__ V, const float* __restrict__ alpha,
                               const float* __restrict__ denom, const float* __restrict__ xin,
                               const float* __restrict__ dis, const float* __restrict__ ew,
                               float* __restrict__ outAcc, float* __restrict__ agg,
                               int E, int fi) {
  int gw = blockIdx.x * 8 + (threadIdx.x >> 5);
  int lane = threadIdx.x & 31;
  if (gw >= E) return;
  int s = src[gw], d = dst[gw];
  int h = lane >> 4;
  float w = alpha[(size_t)gw * 2 + h] / (denom[(size_t)d * 2 + h] + 1e-16f);
  float e0 = 0.f, e1 = 0.f, e2 = 0.f, e3 = 0.f;
  const float* eap = ea + (size_t)gw * 32;
#pragma unroll
  for (int f = 0; f < 32; ++f) {
    float a = eap[f];
    const float* wp = We + f * HCc + lane * 4;
    e0 += a * wp[0]; e1 += a * wp[1]; e2 += a * wp[2]; e3 += a * wp[3];
  }
  const float* vp = V + (size_t)s * HCc + lane * 4;
  float* op = outAcc + (size_t)d * HCc + lane * 4;
  atomicAdd(op + 0, (vp[0] + e0) * w);
  atomicAdd(op + 1, (vp[1] + e1) * w);
  atomicAdd(op + 2, (vp[2] + e2) * w);
  atomicAdd(op + 3, (vp[3] + e3) * w);
  // degree-normalized linear aggregation (custom conv part)
  float norm = dis[s] * ew[gw] * dis[d];
  int per = fi >> 5;
  for (int j = 0; j < per; ++j) {
    int f = lane + j * 32;
    atomicAdd(agg + (size_t)d * fi + f, norm * xin[(size_t)s * fi + f]);
  }
}

// ---------------- batchnorm (training mode, biased var) ----------------
__global__ void k_bn_stats(const float* __restrict__ h, float* mean, float* var, int n) {
  int c = blockIdx.x;                        // 64 channels
  float s = 0.f, s2 = 0.f;
  for (int i = threadIdx.x; i < n; i += blockDim.x) {
    float v = h[(size_t)i * EMBc + c];
    s += v; s2 += v * v;
  }
  __shared__ float sh[256], sh2[256];
  sh[threadIdx.x] = s; sh2[threadIdx.x] = s2;
  __syncthreads();
  for (int off = 128; off; off >>= 1) {
    if (threadIdx.x < off) { sh[threadIdx.x] += sh[threadIdx.x + off];
                             sh2[threadIdx.x] += sh2[threadIdx.x + off]; }
    __syncthreads();
  }
  if (threadIdx.x == 0) {
    float mu = sh[0] / n;
    mean[c] = mu;
    var[c] = sh2[0] / n - mu * mu;
  }
}
__global__ void k_bn_apply(float* __restrict__ h, const float* __restrict__ mean,
                           const float* __restrict__ var, const float* __restrict__ g,
                           const float* __restrict__ b, int total) {
  int i = blockIdx.x * blockDim.x + threadIdx.x;
  if (i >= total) return;
  int c = i & (EMBc - 1);
  h[i] = (h[i] - mean[c]) * rsqrtf(var[c] + 1e-5f) * g[c] + b[c];
}

// ---------------- pooling ----------------
__global__ void k_wnorm(const float* __restrict__ w, float* wn) {
  __shared__ float s[64];
  int t = threadIdx.x;
  s[t] = w[t] * w[t];
  __syncthreads();
  for (int off = 32; off; off >>= 1) { if (t < off) s[t] += s[t + off]; __syncthreads(); }
  if (t == 0) wn[0] = sqrtf(s[0]);
}
__global__ void k_scores(const float* __restrict__ h, const float* __restrict__ w,
                         const float* __restrict__ wn, float* __restrict__ scores, int n) {
  int i = blockIdx.x * blockDim.x + threadIdx.x;
  if (i >= n) return;
  float acc = 0.f;
#pragma unroll
  for (int c = 0; c < EMBc; ++c) acc += h[(size_t)i * EMBc + c] * w[c];
  scores[i] = acc / wn[0];
}
// per-graph block: bitonic sort 1024 scores desc, take top 512, [gmp || gap] of x*tanh(s)
__global__ void k_topk(const float* __restrict__ h, const float* __restrict__ scores,
                       float* __restrict__ out) {
  __shared__ float sv[NPGc];
  __shared__ int   si[NPGc];
  __shared__ float rmx[512], rsm[512];
  int b = blockIdx.x, tid = threadIdx.x;     // 512 threads
  for (int i = tid; i < NPGc; i += 512) { sv[i] = scores[b * NPGc + i]; si[i] = i; }
  __syncthreads();
  for (int ksz = 2; ksz <= NPGc; ksz <<= 1) {
    for (int jj = ksz >> 1; jj > 0; jj >>= 1) {
      for (int t = tid; t < NPGc; t += 512) {
        int p = t ^ jj;
        if (p > t) {
          bool desc = ((t & ksz) == 0);
          float a = sv[t], c = sv[p];
          bool sw = desc ? (a < c) : (a > c);
          if (sw) { sv[t] = c; sv[p] = a; int ti = si[t]; si[t] = si[p]; si[p] = ti; }
        }
      }
      __syncthreads();
    }
  }
  int c = tid & (EMBc - 1);
  int grp = tid >> 6;                        // 8 groups of 64 channels
  float mx = -3.4e38f, sm = 0.f;
  for (int j = grp; j < KKc; j += 8) {
    float t = tanhf(sv[j]);
    float v = h[((size_t)b * NPGc + si[j]) * EMBc + c] * t;
    mx = fmaxf(mx, v); sm += v;
  }
  rmx[tid] = mx; rsm[tid] = sm;
  __syncthreads();
  for (int off = 256; off >= 64; off >>= 1) {
    if (tid < off) { rmx[tid] = fmaxf(rmx[tid], rmx[tid + off]); rsm[tid] += rsm[tid + off]; }
    __syncthreads();
  }
  if (tid < 64) {
    out[b * 2 * EMBc + tid]        = rmx[tid];
    out[b * 2 * EMBc + EMBc + tid] = rsm[tid] * (1.0f / KKc);
  }
}

extern "C" void kernel_launch(void* const* d_in, const int* in_sizes, int n_in,
                              void* d_out, int out_size, void* d_ws, size_t ws_size,
                              hipStream_t stream) {
  const float* x  = (const float*)d_in[0];
  const float* ea = (const float*)d_in[1];
  const float* ew = (const float*)d_in[2];
  const float* P  = (const float*)d_in[3];
  const int*   ei = (const int*)d_in[4];
  const int* src = ei;
  const int* dst = ei + Ee;
  float* out = (float*)d_out;

  // workspace carve (~97 MB total)
  float* base = (float*)d_ws;
  size_t o = 0;
  auto alloc = [&](size_t n) { float* p = base + o; o += n; return p; };
  float*    deg   = alloc(Nn);
  float*    dis   = alloc(Nn);
  float*    nodeQ = alloc((size_t)Nn * HCc);
  float*    nodeK = alloc((size_t)Nn * HCc);
  float*    nodeV = alloc((size_t)Nn * HCc);
  float*    agg   = alloc((size_t)Nn * HCc);
  float*    convO = alloc((size_t)Nn * HCc);
  float*    hbuf  = alloc((size_t)Nn * EMBc);
  float*    alphaA= alloc((size_t)Ee * 2);
  unsigned* mEnc  = (unsigned*)alloc((size_t)Nn * 2);
  float*    denomA= alloc((size_t)Nn * 2);
  float*    meanA = alloc(64);
  float*    varA  = alloc(64);
  float*    wn    = alloc(1);
  float*    scores= alloc(Nn);
  (void)ws_size; (void)in_sizes; (void)n_in; (void)out_size;

  auto cdiv = [](int a, int b) { return (a + b - 1) / b; };
  auto gemm = [&](const float* A, const float* W, const float* bias, float* C,
                  int M, int Ka, int Nc, int flags) {
    dim3 g(Nc / 16, M / 128);
    k_gemm_wmma<<<g, 256, 0, stream>>>(A, W, bias, C, M, Ka, Nc, flags);
  };

  // degree normalization (shared by both conv layers)
  k_fill_f32<<<cdiv(Nn, 256), 256, 0, stream>>>(deg, 0.f, Nn);
  k_deg<<<cdiv(Ee, 256), 256, 0, stream>>>(src, deg, Ee);
  k_dis<<<cdiv(Nn, 256), 256, 0, stream>>>(deg, dis, Nn);

  // ================= layer 1 (fi = 128) =================
  gemm(x, P + C1_WQ, P + C1_BQ, nodeQ, Nn, 128, HCc, 0);
  gemm(x, P + C1_WK, P + C1_BK, nodeK, Nn, 128, HCc, 0);
  gemm(x, P + C1_WV, P + C1_BV, nodeV, Nn, 128, HCc, 0);
  gemm(x, P + C1_WSKIP, P + C1_BSKIP, convO, Nn, 128, HCc, 0);   // out starts as skip
  k_fill_u32<<<cdiv(Nn * 2, 256), 256, 0, stream>>>(mEnc, ENC_NEG_INF, Nn * 2);
  k_fill_f32<<<cdiv(Nn * 2, 256), 256, 0, stream>>>(denomA, 0.f, Nn * 2);
  k_fill_f32<<<cdiv(Nn * 128, 256), 256, 0, stream>>>(agg, 0.f, Nn * 128);
  k_edge_alpha<<<Ee / 8, 256, 0, stream>>>(src, dst, ea, P + C1_WE, nodeQ, nodeK,
                                           alphaA, mEnc, Ee);
  k_edge_exp<<<cdiv(Ee, 256), 256, 0, stream>>>(dst, alphaA, mEnc, denomA, Ee);
  k_edge_scatter<<<Ee / 8, 256, 0, stream>>>(src, dst, ea, P + C1_WE, nodeV, alphaA,
                                             denomA, x, dis, ew, convO, agg, Ee, 128);
  gemm(agg, P + C1_WWL, P + C1_BWL, convO, Nn, 128, HCc, 1);     // += agg @ Wwl + bwl
  gemm(convO, P + T1_W, P + T1_B, hbuf, Nn, HCc, EMBc, 2);       // relu(transf1)
  k_bn_stats<<<64, 256, 0, stream>>>(hbuf, meanA, varA, Nn);
  k_bn_apply<<<cdiv(Nn * EMBc, 256), 256, 0, stream>>>(hbuf, meanA, varA,
                                                       P + BN1_G, P + BN1_B, Nn * EMBc);

  // ================= layer 2 (fi = 64) =================
  gemm(hbuf, P + C2_WQ, P + C2_BQ, nodeQ, Nn, 64, HCc, 0);
  gemm(hbuf, P + C2_WK, P + C2_BK, nodeK, Nn, 64, HCc, 0);
  gemm(hbuf, P + C2_WV, P + C2_BV, nodeV, Nn, 64, HCc, 0);
  gemm(hbuf, P + C2_WSKIP, P + C2_BSKIP, convO, Nn, 64, HCc, 0);
  k_fill_u32<<<cdiv(Nn * 2, 256), 256, 0, stream>>>(mEnc, ENC_NEG_INF, Nn * 2);
  k_fill_f32<<<cdiv(Nn * 2, 256), 256, 0, stream>>>(denomA, 0.f, Nn * 2);
  k_fill_f32<<<cdiv(Nn * 64, 256), 256, 0, stream>>>(agg, 0.f, Nn * 64);
  k_edge_alpha<<<Ee / 8, 256, 0, stream>>>(src, dst, ea, P + C2_WE, nodeQ, nodeK,
                                           alphaA, mEnc, Ee);
  k_edge_exp<<<cdiv(Ee, 256), 256, 0, stream>>>(dst, alphaA, mEnc, denomA, Ee);
  k_edge_scatter<<<Ee / 8, 256, 0, stream>>>(src, dst, ea, P + C2_WE, nodeV, alphaA,
                                             denomA, hbuf, dis, ew, convO, agg, Ee, 64);
  gemm(agg, P + C2_WWL, P + C2_BWL, convO, Nn, 64, HCc, 1);
  gemm(convO, P + T2_W, P + T2_B, hbuf, Nn, HCc, EMBc, 2);       // overwrites h (safe)
  k_bn_stats<<<64, 256, 0, stream>>>(hbuf, meanA, varA, Nn);
  k_bn_apply<<<cdiv(Nn * EMBc, 256), 256, 0, stream>>>(hbuf, meanA, varA,
                                                       P + BN2_G, P + BN2_B, Nn * EMBc);

  // ================= top-k pooling =================
  k_wnorm<<<1, 64, 0, stream>>>(P + POOL_W, wn);
  k_scores<<<cdiv(Nn, 256), 256, 0, stream>>>(hbuf, P + POOL_W, wn, scores, Nn);
  k_topk<<<Bb, 512, 0, stream>>>(hbuf, scores, out);
}